// EGCN2_69217692942494
// MI455X (gfx1250) — compile-verified
//
#include <hip/hip_runtime.h>
#include <hip/hip_bf16.h>
#include <math.h>

// Problem constants (from reference)
#define NN    50000
#define NNPAD 50048            // ceil(NN/64)*64 -> unguarded GEMM stores
#define EE    400000
#define FIN   128
#define HH    256
#define BN_EPS 1e-5f

typedef __attribute__((ext_vector_type(2))) float v2f;
typedef __attribute__((ext_vector_type(8))) float v8f;

// ---------------------------------------------------------------------------
// vectorized fill (all our sizes are multiples of 4 floats)
// ---------------------------------------------------------------------------
__global__ void fill4_kernel(float4* __restrict__ p, float v, size_t n4) {
    size_t i = (size_t)blockIdx.x * blockDim.x + threadIdx.x;
    size_t stride = (size_t)gridDim.x * blockDim.x;
    float4 val = make_float4(v, v, v, v);
    for (; i < n4; i += stride) p[i] = val;
}

// ---------------------------------------------------------------------------
// degree scatter: deg[dst[e]] += 1  (deg pre-filled with 1.0 for self loop)
// ---------------------------------------------------------------------------
__global__ void degree_kernel(const int* __restrict__ ei, float* __restrict__ deg, int E) {
    int e = blockIdx.x * blockDim.x + threadIdx.x;
    if (e < E) atomicAdd(&deg[ei[E + e]], 1.0f);
}

__global__ void rsqrt_inplace_kernel(float* __restrict__ p, int n) {
    int i = blockIdx.x * blockDim.x + threadIdx.x;
    if (i < n) p[i] = rsqrtf(p[i]);
}

// ---------------------------------------------------------------------------
// FP32 WMMA GEMM:  C[M x NC] = A[M x K] @ B[K x NC]
// Block: 256 threads = 8 waves; block tile 64(M) x 32(N); K chunk 32.
// Wave grid 4x2, each wave owns one 16x16 tile via V_WMMA_F32_16X16X4_F32.
// - Branch-free clamped b128 global loads, b128 LDS stores
// - Register double-buffering of the next K chunk
// - All 16 fragment LDS loads hoisted, then 8 back-to-back WMMAs
// Requires: K % 32 == 0, NC % 32 == 0; C must have >= ceil(M/64)*64 rows.
// ---------------------------------------------------------------------------
#define TM 64
#define TN 32
#define TK 32
#define AS_STR 36   // 36*i mod 64 distinct for i=0..15 -> conflict-free A reads
#define BS_STR 40   // (+2 rows) ≡ +16 banks -> half-waves hit disjoint banks

__global__ __launch_bounds__(256)
void gemm_f32_wmma(const float* __restrict__ A, const float* __restrict__ B,
                   float* __restrict__ C, int M, int K, int NC) {
    __shared__ float As[TM * AS_STR];   // 9216 B
    __shared__ float Bs[TK * BS_STR];   // 5120 B

    const int t    = threadIdx.x;
    const int lane = t & 31;
    const int wave = t >> 5;
    const int wm   = wave & 3;
    const int wn   = wave >> 2;
    const int half = lane >> 4;
    const int lp   = lane & 15;
    const int mBase = blockIdx.x * TM;
    const int nBase = blockIdx.y * TN;

    // cooperative-load mapping: A = 2 float4/thread, B = 1 float4/thread
    const int ar  = t >> 3;            // 0..31 (and +32)
    const int ac4 = (t & 7) * 4;       // 0,4,...,28
    const int gr0 = mBase + ar;
    const int gr1 = mBase + ar + 32;
    const int cr0 = gr0 < M ? gr0 : M - 1;   // clamp -> unconditional load
    const int cr1 = gr1 < M ? gr1 : M - 1;
    const bool ok0 = gr0 < M, ok1 = gr1 < M;

    float4 rA0, rA1, rB;
    const float4 z4 = make_float4(0.f, 0.f, 0.f, 0.f);

    // prologue: load K-chunk 0 into registers
    rA0 = *(const float4*)(A + (size_t)cr0 * K + ac4);
    rA1 = *(const float4*)(A + (size_t)cr1 * K + ac4);
    if (!ok0) rA0 = z4;
    if (!ok1) rA1 = z4;
    rB  = *(const float4*)(B + (size_t)ar * NC + nBase + ac4);

    v8f acc = {0.f, 0.f, 0.f, 0.f, 0.f, 0.f, 0.f, 0.f};

    for (int k0 = 0; k0 < K; k0 += TK) {
        // commit current chunk to LDS
        *(float4*)&As[ar * AS_STR + ac4]        = rA0;
        *(float4*)&As[(ar + 32) * AS_STR + ac4] = rA1;
        *(float4*)&Bs[ar * BS_STR + ac4]        = rB;
        __syncthreads();

        // prefetch next chunk into registers (overlaps with WMMA below)
        int k1 = k0 + TK;
        if (k1 < K) {
            rA0 = *(const float4*)(A + (size_t)cr0 * K + k1 + ac4);
            rA1 = *(const float4*)(A + (size_t)cr1 * K + k1 + ac4);
            if (!ok0) rA0 = z4;
            if (!ok1) rA1 = z4;
            rB  = *(const float4*)(B + (size_t)(k1 + ar) * NC + nBase + ac4);
        }

        // ISA 7.12.2 fragment layouts (32-bit A 16x4 / B 4x16):
        //   A: lane lp = row M; VGPR j = K (kk + 2*half + j)
        //   B: VGPR j = row K (kk + 2*half + j), N = lp per half-wave
        const float* ap = As + (wm * 16 + lp) * AS_STR + 2 * half;
        const float* bp = Bs + (2 * half) * BS_STR + wn * 16 + lp;
        v2f af[8], bf[8];
#pragma unroll
        for (int j = 0; j < 8; ++j) {
            af[j].x = ap[4 * j];
            af[j].y = ap[4 * j + 1];
            bf[j].x = bp[(4 * j) * BS_STR];
            bf[j].y = bp[(4 * j + 1) * BS_STR];
        }
#pragma unroll
        for (int j = 0; j < 8; ++j)
            acc = __builtin_amdgcn_wmma_f32_16x16x4_f32(
                false, af[j], false, bf[j], (short)0, acc, false, false);
        __syncthreads();
    }

    // D layout: VGPR r -> (M = r + 8*half, N = lp); C is row-padded, no guard
    const int col = nBase + wn * 16 + lp;
#pragma unroll
    for (int r = 0; r < 8; ++r)
        C[(size_t)(mBase + wm * 16 + 8 * half + r) * NC + col] = acc[r];
}

// ---------------------------------------------------------------------------
// GCN aggregation: agg[dst] += feat[src] * (dinv[src]*dinv[dst])
// over E edges + N self loops. One wave per edge, lane owns 8 channels.
// ---------------------------------------------------------------------------
__global__ void aggregate_kernel(const float* __restrict__ feat, float* __restrict__ agg,
                                 const int* __restrict__ ei, const float* __restrict__ dinv,
                                 int E, int Nn) {
    unsigned wid  = (blockIdx.x * blockDim.x + threadIdx.x) >> 5;
    unsigned lane = threadIdx.x & 31;
    if (wid >= (unsigned)(E + Nn)) return;

    int s, d; float nrm;
    if (wid < (unsigned)E) {
        s = ei[wid]; d = ei[E + wid];
        nrm = dinv[s] * dinv[d];
    } else {
        s = d = (int)(wid - (unsigned)E);
        float r = dinv[s];
        nrm = r * r;
    }
    const float4* pf = (const float4*)(feat + (size_t)s * HH + lane * 8);
    float4 v0 = pf[0], v1 = pf[1];
    float* pa = agg + (size_t)d * HH + lane * 8;
    atomicAdd(pa + 0, v0.x * nrm); atomicAdd(pa + 1, v0.y * nrm);
    atomicAdd(pa + 2, v0.z * nrm); atomicAdd(pa + 3, v0.w * nrm);
    atomicAdd(pa + 4, v1.x * nrm); atomicAdd(pa + 5, v1.y * nrm);
    atomicAdd(pa + 6, v1.z * nrm); atomicAdd(pa + 7, v1.w * nrm);
}

// ---------------------------------------------------------------------------
// BatchNorm stats: per-channel sum / sumsq of (agg + b1). Thread = channel ->
// fully coalesced row reads; per-block partials merged with 2 atomics/channel.
// ---------------------------------------------------------------------------
__global__ void bn_partial_kernel(const float* __restrict__ buf, const float* __restrict__ b1,
                                  float* __restrict__ stats, int Nn) {
    int c = threadIdx.x;                  // 256 threads = channels
    float bias = b1[c];
    float s = 0.f, ss = 0.f;
    for (int r = blockIdx.x; r < Nn; r += gridDim.x) {
        float v = buf[(size_t)r * HH + c] + bias;
        s += v; ss += v * v;
    }
    atomicAdd(&stats[c], s);
    atomicAdd(&stats[HH + c], ss);
}

__global__ void bn_final_kernel(float* __restrict__ stats, int Nn) {
    int c = threadIdx.x;                  // 1 block, 256 threads
    float inv_n = 1.0f / (float)Nn;
    float mu  = stats[c] * inv_n;
    float var = stats[HH + c] * inv_n - mu * mu;   // biased variance
    stats[2 * HH + c] = mu;
    stats[3 * HH + c] = rsqrtf(var + BN_EPS);
}

__global__ void bn_apply_relu_kernel(float* __restrict__ buf, const float* __restrict__ b1,
                                     const float* __restrict__ gamma, const float* __restrict__ beta,
                                     const float* __restrict__ stats, size_t total) {
    size_t i = (size_t)blockIdx.x * blockDim.x + threadIdx.x;
    if (i >= total) return;
    int c = (int)(i & (HH - 1));
    float v = buf[i] + b1[c];
    float h = (v - stats[2 * HH + c]) * stats[3 * HH + c] * gamma[c] + beta[c];
    buf[i] = fmaxf(h, 0.0f);
}

__global__ void bias_relu_kernel(float* __restrict__ buf, const float* __restrict__ b, size_t total) {
    size_t i = (size_t)blockIdx.x * blockDim.x + threadIdx.x;
    if (i >= total) return;
    int c = (int)(i & (HH - 1));
    buf[i] = fmaxf(buf[i] + b[c], 0.0f);
}

// ---------------------------------------------------------------------------
// Edge MLP: out[e] = sigmoid( relu(P[src]+Q[dst]+bm1) . Wm2 + bm2 )
// One wave32 per edge; lane owns 8 channels; shfl_xor tree reduction.
// P,Q (100 MB) are L2-resident -> gathers served from L2.
// ---------------------------------------------------------------------------
__global__ void edge_mlp_kernel(const float* __restrict__ P, const float* __restrict__ Q,
                                const int* __restrict__ ei, const float* __restrict__ bm1,
                                const float* __restrict__ Wm2, const float* __restrict__ bm2,
                                float* __restrict__ out, int E) {
    unsigned wid  = (blockIdx.x * blockDim.x + threadIdx.x) >> 5;
    unsigned lane = threadIdx.x & 31;
    if (wid >= (unsigned)E) return;
    int s = ei[wid], d = ei[E + wid];

    const float4* ps = (const float4*)(P + (size_t)s * HH + lane * 8);
    const float4* pd = (const float4*)(Q + (size_t)d * HH + lane * 8);
    const float4* pb = (const float4*)(bm1 + lane * 8);
    const float4* pw = (const float4*)(Wm2 + lane * 8);

    float acc = 0.f;
#pragma unroll
    for (int i = 0; i < 2; ++i) {
        float4 a = ps[i], b = pd[i], bb = pb[i], w = pw[i];
        acc += fmaxf(a.x + b.x + bb.x, 0.f) * w.x;
        acc += fmaxf(a.y + b.y + bb.y, 0.f) * w.y;
        acc += fmaxf(a.z + b.z + bb.z, 0.f) * w.z;
        acc += fmaxf(a.w + b.w + bb.w, 0.f) * w.w;
    }
#pragma unroll
    for (int off = 16; off > 0; off >>= 1)
        acc += __shfl_xor(acc, off, 32);

    if (lane == 0) {
        float z = acc + bm2[0];
        out[wid] = 1.0f / (1.0f + __expf(-z));
    }
}

// ---------------------------------------------------------------------------
// launch
// ---------------------------------------------------------------------------
extern "C" void kernel_launch(void* const* d_in, const int* in_sizes, int n_in,
                              void* d_out, int out_size, void* d_ws, size_t ws_size,
                              hipStream_t stream) {
    const float* x     = (const float*)d_in[0];
    const int*   ei    = (const int*)d_in[1];   // JAX x64-off: int32
    const float* W1    = (const float*)d_in[2];
    const float* b1    = (const float*)d_in[3];
    const float* gamma = (const float*)d_in[4];
    const float* beta  = (const float*)d_in[5];
    const float* W2    = (const float*)d_in[6];
    const float* b2    = (const float*)d_in[7];
    const float* Wm1   = (const float*)d_in[8];  // [2H, H] row-major
    const float* bm1   = (const float*)d_in[9];
    const float* Wm2   = (const float*)d_in[10]; // [H, 1]
    const float* bm2   = (const float*)d_in[11];
    float* out = (float*)d_out;

    const size_t NHf    = (size_t)NN * HH;       // live feature region
    const size_t NHpad  = (size_t)NNPAD * HH;    // padded rows for GEMM stores
    float* ws    = (float*)d_ws;
    float* deg   = ws;                           // NN floats (becomes dinv)
    float* stats = ws + NN;                      // 4*HH: sum, sumsq, mu, rsig
    float* bufA  = stats + 4 * HH;               // NHpad
    float* bufB  = bufA + NHpad;                 // NHpad
    float* bufC  = bufB + NHpad;                 // NHpad  (total ~154 MB)

    const int T = 256;
    const dim3 gGemm(NNPAD / TM, HH / TN);

    // 1) degree (A + I) and dinv
    fill4_kernel<<<64, T, 0, stream>>>((float4*)deg, 1.0f, (size_t)NN / 4);
    degree_kernel<<<(EE + T - 1) / T, T, 0, stream>>>(ei, deg, EE);
    rsqrt_inplace_kernel<<<(NN + T - 1) / T, T, 0, stream>>>(deg, NN);

    // 2) t0 = x @ W1   [N,128]x[128,256]
    gemm_f32_wmma<<<gGemm, T, 0, stream>>>(x, W1, bufA, NN, FIN, HH);

    // 3) agg1 = scatter(t0)  -> bufB
    fill4_kernel<<<2048, T, 0, stream>>>((float4*)bufB, 0.0f, NHf / 4);
    fill4_kernel<<<2, T, 0, stream>>>((float4*)stats, 0.0f, (size_t)HH);
    {
        long long thr = (long long)(EE + NN) * 32;
        aggregate_kernel<<<(int)((thr + T - 1) / T), T, 0, stream>>>(bufA, bufB, ei, deg, EE, NN);
    }

    // 4) batchnorm (batch stats on agg1 + b1) + relu  -> h1 in bufB
    bn_partial_kernel<<<512, HH, 0, stream>>>(bufB, b1, stats, NN);
    bn_final_kernel<<<1, HH, 0, stream>>>(stats, NN);
    bn_apply_relu_kernel<<<(int)((NHf + T - 1) / T), T, 0, stream>>>(bufB, b1, gamma, beta, stats, NHf);

    // 5) t1 = h1 @ W2  -> bufA
    gemm_f32_wmma<<<gGemm, T, 0, stream>>>(bufB, W2, bufA, NN, HH, HH);

    // 6) agg2 = scatter(t1) -> bufB ; h2 = relu(agg2 + b2)
    fill4_kernel<<<2048, T, 0, stream>>>((float4*)bufB, 0.0f, NHf / 4);
    {
        long long thr = (long long)(EE + NN) * 32;
        aggregate_kernel<<<(int)((thr + T - 1) / T), T, 0, stream>>>(bufA, bufB, ei, deg, EE, NN);
    }
    bias_relu_kernel<<<(int)((NHf + T - 1) / T), T, 0, stream>>>(bufB, b2, NHf);

    // 7) factor edge-MLP layer 1 through nodes:
    //    P = h2 @ Wm1[0:H,:] -> bufA ;  Q = h2 @ Wm1[H:2H,:] -> bufC
    gemm_f32_wmma<<<gGemm, T, 0, stream>>>(bufB, Wm1, bufA, NN, HH, HH);
    gemm_f32_wmma<<<gGemm, T, 0, stream>>>(bufB, Wm1 + (size_t)HH * HH, bufC, NN, HH, HH);

    // 8) per-edge: sigmoid(relu(P[s]+Q[d]+bm1) . Wm2 + bm2)
    {
        long long thr = (long long)EE * 32;
        edge_mlp_kernel<<<(int)((thr + T - 1) / T), T, 0, stream>>>(bufA, bufC, ei, bm1, Wm2, bm2, out, EE);
    }
}